// Model_86586540687789
// MI455X (gfx1250) — compile-verified
//
#include <hip/hip_runtime.h>
#include <stdint.h>

#define DIM       4096
#define TOTAL     16384
#define WIDTH     4
#define STATE_LEN 3
#define BATCH     8
#define NSLOTS    256

#define TILE_T    1024
#define BLOCK     256

typedef float f32x4 __attribute__((ext_vector_type(4)));
typedef int   i32x4v __attribute__((vector_size(16)));   // matches builtin's V4i

#if defined(__HIP_DEVICE_COMPILE__) && __has_builtin(__builtin_amdgcn_global_load_async_to_lds_b128)
#define CDNA5_ASYNC 1
#endif

// Stage 16 bytes global -> LDS. CDNA5 path: async DMA tracked by ASYNCcnt.
__device__ __forceinline__ void stage16(const float* g, float* l) {
#if defined(CDNA5_ASYNC)
    __builtin_amdgcn_global_load_async_to_lds_b128(
        (__attribute__((address_space(1))) i32x4v*)g,
        (__attribute__((address_space(3))) i32x4v*)l,
        /*imm offset*/0, /*cpol*/0);
#else
    *(f32x4*)l = *(const f32x4*)g;
#endif
}

__device__ __forceinline__ void stage_wait() {
#if defined(CDNA5_ASYNC)
#if __has_builtin(__builtin_amdgcn_s_wait_asynccnt)
    __builtin_amdgcn_s_wait_asynccnt(0);
#else
    asm volatile("s_wait_asynccnt 0" ::: "memory");
#endif
#endif
}

// ---------------------------------------------------------------------------
// Kernel 1: causal depthwise conv (width 4) + residual.
// Block: (row d) x (1024-wide t tile). 256 threads, 4 outputs each.
// ---------------------------------------------------------------------------
__global__ __launch_bounds__(BLOCK) void conv1d_kernel(
    const float* __restrict__ x, const float* __restrict__ weight,
    const float* __restrict__ conv_states, const int* __restrict__ qsl,
    const int* __restrict__ cache_indices, const int* __restrict__ init_mode,
    const int* __restrict__ pad_slot_id, const int* __restrict__ resid,
    float* __restrict__ out)
{
    __shared__ __align__(16) float smem[TILE_T + 16]; // [0..3]=halo, [4..]=tile
    __shared__ int s_qsl[BATCH + 1];
    __shared__ int s_ci[BATCH];
    __shared__ int s_im[BATCH];

    const int tid   = threadIdx.x;
    const int tile0 = blockIdx.x * TILE_T;
    const int d     = blockIdx.y;
    const float* xrow = x + (size_t)d * TOTAL;

    if (tid < BATCH + 1) s_qsl[tid] = qsl[tid];
    if (tid >= 32 && tid < 32 + BATCH) {
        s_ci[tid - 32] = cache_indices[tid - 32];
        s_im[tid - 32] = init_mode[tid - 32];
    }

    // Async-stage tile (+halo) into LDS: 16 B per lane.
    stage16(xrow + tile0 + tid * 4, &smem[4 + tid * 4]);
    if (tid == 0) {
        const float* gh = (tile0 >= 4) ? (xrow + tile0 - 4) : xrow; // clamp; unused if tile0==0
        stage16(gh, &smem[0]);
    }
    stage_wait();
    __syncthreads();

    const int t0 = tile0 + tid * 4;
    f32x4 lo = *(const f32x4*)&smem[tid * 4];     // x[t0-4 .. t0-1]
    f32x4 hi = *(const f32x4*)&smem[tid * 4 + 4]; // x[t0   .. t0+3]
    float lx[8] = { lo[0], lo[1], lo[2], lo[3], hi[0], hi[1], hi[2], hi[3] };

    const f32x4 w   = *(const f32x4*)(weight + (size_t)d * WIDTH); // w[k], k=0..3
    const int rflag = *resid;
    const int pad   = *pad_slot_id;

    int seg0 = 0, seg3 = 0;
#pragma unroll
    for (int b = 1; b < BATCH; ++b) {
        seg0 += (s_qsl[b] <= t0);
        seg3 += (s_qsl[b] <= t0 + 3);
    }

    f32x4 o;
    if (seg0 == seg3 && (t0 - s_qsl[seg0]) >= STATE_LEN) {
        // Fast path: all taps in-sequence -> pure FMA chain.
#pragma unroll
        for (int j = 0; j < 4; ++j) {
            float v = w[0] * lx[j + 1] + w[1] * lx[j + 2]
                    + w[2] * lx[j + 3] + w[3] * lx[j + 4];
            if (rflag) v += lx[j + 4];
            o[j] = v;
        }
    } else {
        // Boundary path: per element, taps may come from the cached state.
#pragma unroll
        for (int j = 0; j < 4; ++j) {
            const int t = t0 + j;
            int seg = 0;
#pragma unroll
            for (int b = 1; b < BATCH; ++b) seg += (s_qsl[b] <= t);
            const int  p       = t - s_qsl[seg];
            const int  rawSlot = s_ci[seg];
            const bool valid   = (rawSlot != pad);
            const int  slot    = valid ? rawSlot : 0;
            const bool useInit = valid && (s_im[seg] != 0);
            const float* st = conv_states + ((size_t)slot * DIM + d) * STATE_LEN;

            float v = w[3] * lx[j + 4];
#pragma unroll
            for (int dd = 1; dd <= 3; ++dd) {       // tap distance; weight k = 3-dd
                const float wk = (dd == 1) ? w[2] : (dd == 2) ? w[1] : w[0];
                float val;
                if (p >= dd)       val = lx[j + 4 - dd];
                else if (useInit)  val = st[STATE_LEN - dd + p]; // in [0,2] since p<dd
                else               val = 0.0f;
                v += wk * val;
            }
            if (rflag) v += lx[j + 4];
            o[j] = v;
        }
    }

    // out is never re-read: non-temporal 128-bit store.
    __builtin_nontemporal_store(o, (f32x4*)(out + (size_t)d * TOTAL + t0));
}

// ---------------------------------------------------------------------------
// Kernel 2: produce updated conv_states (copy + scatter-replace <=8 slots).
// One thread per output element over NSLOTS*DIM*STATE_LEN.
// ---------------------------------------------------------------------------
__global__ __launch_bounds__(256) void state_kernel(
    const float* __restrict__ x, const float* __restrict__ conv_states,
    const int* __restrict__ qsl, const int* __restrict__ cache_indices,
    const int* __restrict__ init_mode, const int* __restrict__ pad_slot_id,
    float* __restrict__ out_states)
{
    const int n = NSLOTS * DIM * STATE_LEN;
    const int idx = blockIdx.x * blockDim.x + threadIdx.x;
    if (idx >= n) return;

    const int s = idx / (DIM * STATE_LEN);
    const int r = idx - s * (DIM * STATE_LEN);
    const int d = r / STATE_LEN;
    const int j = r - d * STATE_LEN;
    const int pad = *pad_slot_id;

    int hit = -1;
#pragma unroll
    for (int b = 0; b < BATCH; ++b) {
        const int c = cache_indices[b];
        if (c == s && c != pad) hit = b;   // last writer wins (indices are distinct)
    }

    float v;
    if (hit < 0) {
        v = conv_states[idx];
    } else {
        const int start = qsl[hit];
        const int end   = qsl[hit + 1];
        const int L     = end - start;
        const int gx    = end + j - STATE_LEN;
        if (gx >= start) {
            int g = gx < 0 ? 0 : (gx > TOTAL - 1 ? TOTAL - 1 : gx);
            v = x[(size_t)d * TOTAL + g];
        } else if (init_mode[hit] != 0) {
            int si = L + j;
            si = si < 0 ? 0 : (si > STATE_LEN - 1 ? STATE_LEN - 1 : si);
            v = conv_states[((size_t)s * DIM + d) * STATE_LEN + si];
        } else {
            v = 0.0f;
        }
    }
    out_states[idx] = v;
}

extern "C" void kernel_launch(void* const* d_in, const int* in_sizes, int n_in,
                              void* d_out, int out_size, void* d_ws, size_t ws_size,
                              hipStream_t stream) {
    const float* x           = (const float*)d_in[0];
    const float* weight      = (const float*)d_in[1];
    const float* conv_states = (const float*)d_in[2];
    const int*   qsl         = (const int*)d_in[3];
    const int*   ci          = (const int*)d_in[4];
    const int*   im          = (const int*)d_in[5];
    const int*   pad         = (const int*)d_in[6];
    const int*   resid       = (const int*)d_in[7];

    float* out        = (float*)d_out;
    float* out_states = out + (size_t)DIM * TOTAL;

    dim3 grid1(TOTAL / TILE_T, DIM);
    conv1d_kernel<<<grid1, BLOCK, 0, stream>>>(x, weight, conv_states, qsl, ci,
                                               im, pad, resid, out);

    const int n2 = NSLOTS * DIM * STATE_LEN;
    state_kernel<<<(n2 + 255) / 256, 256, 0, stream>>>(x, conv_states, qsl, ci,
                                                       im, pad, out_states);
}